// ScaledDotProductAttention_7954279432295
// MI455X (gfx1250) — compile-verified
//
#include <hip/hip_runtime.h>

typedef __attribute__((ext_vector_type(2))) float v2f;
typedef __attribute__((ext_vector_type(8))) float v8f;

#define NH 16
#define SEQ 4096
#define DIM 64
#define CHUNKS 16
#define CHUNK_LEN (SEQ / CHUNKS)   // 256

// ---------------------------------------------------------------------------
// Phase 1: part[h][ch] = K_chunk^T @ V_chunk   (64x64 fp32 tile per chunk)
// M-dim = dk, N-dim = dv, K-dim = s (reduced 4 per WMMA, fp32 accum).
// 16 waves/block, one 16x16 tile per wave; grid = NH*CHUNKS.
// ---------------------------------------------------------------------------
__global__ __launch_bounds__(512) void ktv_partial_kernel(
    const float* __restrict__ K, const float* __restrict__ V,
    float* __restrict__ part)
{
  const int h    = blockIdx.x / CHUNKS;
  const int ch   = blockIdx.x % CHUNKS;
  const int wave = threadIdx.x >> 5;
  const int lane = threadIdx.x & 31;
  const int half = lane >> 4;        // 0: K=0,1   1: K=2,3
  const int lr   = lane & 15;
  const int ti   = wave >> 2;        // dk tile (0..3)
  const int tj   = wave & 3;         // dv tile (0..3)
  const int dk   = ti * 16 + lr;     // A: lane = M (= dk)
  const int dv   = tj * 16 + lr;     // B: lane = N (= dv)

  const float* Kh = K + (size_t)h * SEQ * DIM;
  const float* Vh = V + (size_t)h * SEQ * DIM;
  const int s_begin = ch * CHUNK_LEN;

  v8f acc = {};
  for (int s0 = s_begin; s0 < s_begin + CHUNK_LEN; s0 += 4) {
    const int r0 = s0 + 2 * half;
    v2f a, b;
    // A[m][k] = K[h][s0+k][dk]; VGPR0 = K{0,2}, VGPR1 = K{1,3}
    a.x = Kh[(size_t)(r0    ) * DIM + dk];
    a.y = Kh[(size_t)(r0 + 1) * DIM + dk];
    // B[k][n] = V[h][s0+k][dv]; same K striping, lane = N
    b.x = Vh[(size_t)(r0    ) * DIM + dv];
    b.y = Vh[(size_t)(r0 + 1) * DIM + dv];
    acc = __builtin_amdgcn_wmma_f32_16x16x4_f32(
        /*neg_a=*/false, a, /*neg_b=*/false, b,
        /*c_mod=*/(short)0, acc, /*reuse_a=*/false, /*reuse_b=*/false);
  }

  // C/D layout: VGPR r, lane -> (m = r + 8*half, n = lr)
  float* tile = part + (size_t)(h * CHUNKS + ch) * (DIM * DIM)
              + (size_t)(ti * 16) * DIM + tj * 16;
#pragma unroll
  for (int r = 0; r < 8; ++r)
    tile[(size_t)(r + 8 * half) * DIM + lr] = acc[r];
}

// ---------------------------------------------------------------------------
// Phase 2: M[h] = sum_ch part[h][ch]   (deterministic tree-free serial sum)
// ---------------------------------------------------------------------------
__global__ __launch_bounds__(256) void ktv_reduce_kernel(
    const float* __restrict__ part, float* __restrict__ M)
{
  const int h = blockIdx.x;
  for (int e = threadIdx.x; e < DIM * DIM; e += 256) {
    float s = 0.0f;
#pragma unroll
    for (int c = 0; c < CHUNKS; ++c)
      s += part[(size_t)(h * CHUNKS + c) * (DIM * DIM) + e];
    M[(size_t)h * (DIM * DIM) + e] = s;
  }
}

// ---------------------------------------------------------------------------
// Phase 3: O[h] = Q[h] (4096x64) @ M[h] (64x64)
// Each block: 64 Q rows x full 64 cols = 4x4 tiles, one per wave.
// K-loop = 16 WMMAs (K=64 in steps of 4). M[h] is 16 KB -> L0/L2 resident.
// ---------------------------------------------------------------------------
__global__ __launch_bounds__(512) void qm_kernel(
    const float* __restrict__ Q, const float* __restrict__ M,
    float* __restrict__ O)
{
  const int blocks_per_head = SEQ / 64;            // 64
  const int h    = blockIdx.x / blocks_per_head;
  const int sb   = (blockIdx.x % blocks_per_head) * 64;
  const int wave = threadIdx.x >> 5;
  const int lane = threadIdx.x & 31;
  const int half = lane >> 4;
  const int lr   = lane & 15;
  const int ti   = wave >> 2;                      // row tile
  const int tj   = wave & 3;                       // dv tile
  const int row  = sb + ti * 16 + lr;              // A: lane = M (= q row)
  const int dv   = tj * 16 + lr;                   // B: lane = N (= dv)

  const float* Qh = Q + (size_t)h * SEQ * DIM;
  const float* Mh = M + (size_t)h * DIM * DIM;

  v8f acc = {};
#pragma unroll
  for (int k0 = 0; k0 < DIM; k0 += 4) {
    const int kk = k0 + 2 * half;
    v2f a, b;
    // A[m][k] = Q[h][row][k]; k,k+1 contiguous -> pairwise load
    const float* qp = Qh + (size_t)row * DIM + kk;
    a.x = qp[0];
    a.y = qp[1];
    // B[k][n] = M[h][k][dv]
    b.x = Mh[(size_t)(kk    ) * DIM + dv];
    b.y = Mh[(size_t)(kk + 1) * DIM + dv];
    acc = __builtin_amdgcn_wmma_f32_16x16x4_f32(
        false, a, false, b, (short)0, acc, false, false);
  }

  float* op = O + (size_t)h * SEQ * DIM + (size_t)(sb + ti * 16) * DIM + tj * 16;
#pragma unroll
  for (int r = 0; r < 8; ++r)
    op[(size_t)(r + 8 * half) * DIM + lr] = acc[r];
}

// ---------------------------------------------------------------------------
extern "C" void kernel_launch(void* const* d_in, const int* in_sizes, int n_in,
                              void* d_out, int out_size, void* d_ws, size_t ws_size,
                              hipStream_t stream)
{
  const float* q = (const float*)d_in[0];
  const float* k = (const float*)d_in[1];
  const float* v = (const float*)d_in[2];
  float* out = (float*)d_out;

  // workspace layout: [partials: NH*CHUNKS*64*64 f32 = 4 MB][M: NH*64*64 f32]
  float* part = (float*)d_ws;
  float* Mbuf = part + (size_t)NH * CHUNKS * DIM * DIM;

  ktv_partial_kernel<<<NH * CHUNKS, 512, 0, stream>>>(k, v, part);
  ktv_reduce_kernel<<<NH, 256, 0, stream>>>(part, Mbuf);
  qm_kernel<<<NH * (SEQ / 64), 512, 0, stream>>>(q, Mbuf, out);
}